// PeriodicAggregate2D_17841294147819
// MI455X (gfx1250) — compile-verified
//
#include <hip/hip_runtime.h>

// ---------------------------------------------------------------------------
// PeriodicAggregate2D: out[b, k1, k2] = sum over x[b, i, j] with i%8==k1, j%8==k2
// x: (32, 1024, 1024) f32.  out: (32, 8, 8) f32.
// Memory-bound: 128 MiB streamed once -> ~5.8us at 23.3 TB/s.
// ---------------------------------------------------------------------------

#define BATCHES   32
#define H         1024
#define W         1024
#define ROWCHUNKS 16          // blocks per batch along rows
#define ROWS_PB   (H / ROWCHUNKS)   // 64 rows per block
#define TPB       256         // 8 waves; 256*4 floats == 1024 columns exactly

typedef __attribute__((ext_vector_type(2))) float v2f;
typedef __attribute__((ext_vector_type(8))) float v8f;

// Kernel 1: stream x, produce per-block 64-bin partial sums (deterministic).
__global__ __launch_bounds__(TPB) void pa2d_partial(const float* __restrict__ x,
                                                    float* __restrict__ ws) {
    const int rc    = blockIdx.x;   // 0..15 row chunk
    const int batch = blockIdx.y;   // 0..31
    const int t     = threadIdx.x;  // 0..255 -> columns 4t..4t+3

    const float4* xb =
        reinterpret_cast<const float4*>(x + (size_t)batch * (size_t)(H * W));

    // acc[k1][c]: row-residue k1, local column c (actual col residue = (t&1)*4+c)
    float acc[8][4];
    #pragma unroll
    for (int k1 = 0; k1 < 8; ++k1)
        #pragma unroll
        for (int c = 0; c < 4; ++c) acc[k1][c] = 0.0f;

    const int row0 = rc * ROWS_PB;
    for (int g = 0; g < ROWS_PB / 8; ++g) {
        const int r = row0 + g * 8;
        #pragma unroll
        for (int k1 = 0; k1 < 8; ++k1) {
            float4 v = xb[(size_t)(r + k1) * (W / 4) + t];   // global_load_b128
            acc[k1][0] += v.x;
            acc[k1][1] += v.y;
            acc[k1][2] += v.z;
            acc[k1][3] += v.w;
        }
    }

    // Block reduction via LDS, fixed order -> deterministic, bank-conflict free.
    __shared__ float red[TPB * 32];
    #pragma unroll
    for (int k1 = 0; k1 < 8; ++k1)
        #pragma unroll
        for (int c = 0; c < 4; ++c)
            red[t * 32 + k1 * 4 + c] = acc[k1][c];
    __syncthreads();

    if (t < 64) {
        const int k1   = t >> 3;        // row residue
        const int j    = t & 7;         // col residue
        const int half = j >> 2;        // source-thread parity
        const int c    = j & 3;
        float s = 0.0f;
        for (int si = 0; si < TPB / 2; ++si)
            s += red[(2 * si + half) * 32 + k1 * 4 + c];
        ws[((size_t)batch * ROWCHUNKS + rc) * 64 + t] = s;
    }
}

// Kernel 2: per batch, fold 16 partial 64-bin vectors with f32 WMMA.
// D(16x16) = ones(16x4) x P(4x16) + C, chained 4x over rc-chunks.
// A == 1 makes the result exact and invariant to K striping; N = lane&15.
__global__ __launch_bounds__(32) void pa2d_final(const float* __restrict__ ws,
                                                 float* __restrict__ out) {
    const int batch = blockIdx.x;    // 0..31
    const int lane  = threadIdx.x;   // 0..31, one full wave (EXEC all ones)

    const float* P = ws + (size_t)batch * ROWCHUNKS * 64;

    v2f a;                     // A = all-ones 16x4 (every lane holds 1.0,1.0)
    a.x = 1.0f;
    a.y = 1.0f;

    const int n     = lane & 15;   // bin-column within tile (B/D lane mapping)
    const int khalf = lane >> 4;   // which pair of K rows this lane supplies

    #pragma unroll
    for (int btile = 0; btile < 4; ++btile) {   // 4 tiles of 16 bins = 64 bins
        v8f c = {};
        #pragma unroll
        for (int rc0 = 0; rc0 < ROWCHUNKS; rc0 += 4) {
            v2f b;
            b.x = P[(size_t)(rc0 + 2 * khalf + 0) * 64 + btile * 16 + n];
            b.y = P[(size_t)(rc0 + 2 * khalf + 1) * 64 + btile * 16 + n];
            // v_wmma_f32_16x16x4_f32: D = A*B + C
            c = __builtin_amdgcn_wmma_f32_16x16x4_f32(
                    false, a, false, b, (short)0, c, false, false);
        }
        // Every row of D holds the same sum; take M=0 (VGPR0, lanes 0..15).
        if (lane < 16)
            out[(size_t)batch * 64 + btile * 16 + n] = c[0];
    }
}

extern "C" void kernel_launch(void* const* d_in, const int* in_sizes, int n_in,
                              void* d_out, int out_size, void* d_ws, size_t ws_size,
                              hipStream_t stream) {
    (void)in_sizes; (void)n_in; (void)out_size; (void)ws_size;

    const float* x   = (const float*)d_in[0];
    float*       out = (float*)d_out;
    float*       ws  = (float*)d_ws;   // needs 32*16*64*4 = 128 KiB

    dim3 grid1(ROWCHUNKS, BATCHES);
    pa2d_partial<<<grid1, TPB, 0, stream>>>(x, ws);
    pa2d_final<<<BATCHES, 32, 0, stream>>>(ws, out);
}